// RunningAvg_40922448396839
// MI455X (gfx1250) — compile-verified
//
#include <hip/hip_runtime.h>

// Box filter (N=11, 'same') along time of x[8,16384,256] f32.
// Memory-bound: ~268 MB min traffic -> ~11.5us @ 23.3 TB/s.
// Path: GLOBAL_LOAD_ASYNC_TO_LDS_B128 streaming (ASYNCcnt) -> LDS slab with
// zero-padded halo -> banded matmul on V_WMMA_F32_16X16X4_F32.

typedef float v2f __attribute__((ext_vector_type(2)));
typedef float v8f __attribute__((ext_vector_type(8)));

constexpr int BATCH = 8;
constexpr int T     = 16384;
constexpr int C     = 256;
constexpr int NWIN  = 11;
constexpr int HALO  = 5;          // (N-1)/2, symmetric window
constexpr int TB    = 64;         // output time rows per workgroup
constexpr int SLAB_ROWS  = 76;    // rows 0..75: inputs t_blk-5 .. t_blk+70 (K padded to 28)
constexpr int LDS_STRIDE = 272;   // floats; 272%64=16 -> lane-halves hit disjoint banks
constexpr int LDS_BYTES  = SLAB_ROWS * LDS_STRIDE * 4;  // 82,688 B (< 320KB/WGP)

__global__ void __launch_bounds__(256)
box11_wmma_kernel(const float* __restrict__ x, float* __restrict__ out) {
  extern __shared__ float lds[];
  const int tid   = threadIdx.x;
  const int wg    = blockIdx.x;
  const int b     = wg >> 8;             // T/TB = 256 time blocks per batch
  const int t_blk = (wg & 255) * TB;

  // ---------------- Stage 1: async global -> LDS slab fill ----------------
  // Slab covers input times [t_blk-5, t_blk+70], 256 channels, zero halo.
  // Each lane moves 16B (4 channels); 4 rows in flight per loop iteration.
  const int c4 = (tid & 63) << 2;        // channel group, 64 lanes span a 1KB row
  for (int rb = 0; rb < SLAB_ROWS; rb += 4) {
    const int r    = rb + (tid >> 6);    // 0..75
    const int t_in = t_blk - HALO + r;
    const unsigned lds_off = __builtin_amdgcn_groupstaticsize() +
                             (unsigned)((r * LDS_STRIDE + c4) * 4);
    if (t_in >= 0 && t_in < T) {
      const float* gp = x + ((size_t)b * T + t_in) * C + c4;
      asm volatile("global_load_async_to_lds_b128 %0, %1, off"
                   :: "v"(lds_off), "v"(gp) : "memory");
    } else {
      float* lp = &lds[r * LDS_STRIDE + c4];   // zero-pad halo (disjoint rows)
      lp[0] = 0.f; lp[1] = 0.f; lp[2] = 0.f; lp[3] = 0.f;
    }
  }
  asm volatile("s_wait_asynccnt 0x0" ::: "memory");  // my wave's DMA done
  __syncthreads();                                   // everyone's DMA done

  // ------------- Stage 2: banded matmul, V_WMMA_F32_16X16X4_F32 -------------
  // out_tile(16t x 16c) = sum_{j=0..6} A_j(16x4) * B_j(4x16), K window = 26 (pad 28).
  // A[m][q] = 1/11 if q in [m, m+10]   (q indexes input time t0-5+q)
  const int wave  = tid >> 5;
  const int lane  = tid & 31;
  const int m     = lane & 15;          // A: M row | B/D: N column (channel)
  const int khalf = (lane >> 4) << 1;   // 0 or 2: K pair selected by lane half

  v2f aw[7];
  #pragma unroll
  for (int j = 0; j < 7; ++j) {
    const int q0 = 4 * j + khalf;       // VGPR0 holds K=q0, VGPR1 holds K=q0+1
    aw[j].x = (q0     >= m && q0     <= m + (NWIN - 1)) ? (1.0f / NWIN) : 0.0f;
    aw[j].y = (q0 + 1 >= m && q0 + 1 <= m + (NWIN - 1)) ? (1.0f / NWIN) : 0.0f;
  }

  // 4 time-tiles x 16 channel-tiles = 64 tiles; 8 waves -> 8 tiles each.
  for (int i = wave; i < (TB / 16) * (C / 16); i += 8) {
    const int tt = i >> 4;              // time tile 0..3
    const int c0 = (i & 15) << 4;       // channel tile base
    v8f acc = {};
    #pragma unroll
    for (int j = 0; j < 7; ++j) {
      // B 4x16: lane-half picks K pair; lanes 0-15 read 16 contiguous floats.
      const int rbase = (tt << 4) + 4 * j + khalf;    // slab row for K=q0
      const int a0    = rbase * LDS_STRIDE + c0 + m;
      v2f bv;
      bv.x = lds[a0];                   // K = q0
      bv.y = lds[a0 + LDS_STRIDE];      // K = q0+1
      acc = __builtin_amdgcn_wmma_f32_16x16x4_f32(
          /*neg_a=*/false, aw[j], /*neg_b=*/false, bv,
          /*c_mod=*/(short)0, acc, /*reuse_a=*/false, /*reuse_b=*/false);
    }
    // D 16x16: VGPR rr -> M = rr + 8*(lane>=16), N = lane%16.
    const int chan  = c0 + m;
    const int tbase = t_blk + (tt << 4) + ((lane >> 4) << 3);
    #pragma unroll
    for (int rr = 0; rr < 8; ++rr) {
      out[((size_t)b * T + (tbase + rr)) * C + chan] = acc[rr];
    }
  }
}

extern "C" void kernel_launch(void* const* d_in, const int* in_sizes, int n_in,
                              void* d_out, int out_size, void* d_ws, size_t ws_size,
                              hipStream_t stream) {
  (void)in_sizes; (void)n_in; (void)out_size; (void)d_ws; (void)ws_size;
  const float* x = (const float*)d_in[0];
  float* out     = (float*)d_out;
  dim3 grid(BATCH * (T / TB));   // 2048 workgroups
  dim3 block(256);               // 8 wave32
  hipLaunchKernelGGL(box11_wmma_kernel, grid, block, LDS_BYTES, stream, x, out);
}